// GCL4SR_37288906064248
// MI455X (gfx1250) — compile-verified
//
#include <hip/hip_runtime.h>
#include <math.h>

#define Bz   256
#define Lz   50
#define Hz   128
#define NHz  4
#define HDz  32
#define NLz  2
#define FANz 20
#define Tz   (Bz * Lz)      /* 12800 */

typedef float v2f __attribute__((ext_vector_type(2)));
typedef float v8f __attribute__((ext_vector_type(8)));

__device__ __forceinline__ float wave_sum(float v) {
#pragma unroll
  for (int off = 16; off > 0; off >>= 1) v += __shfl_xor(v, off, 32);
  return v;
}

// ---------------------------------------------------------------------------
// fp32 WMMA GEMM: out[M x N] = sum_p A_p (M x K) * W_p(N rows, row-stride ldw)^T
//                 (+ bias[N]) (gelu?) (+ resid[M x N])
// One wave computes a 32x32 output tile (4 accumulators) so each A fragment
// feeds 2 wmmas and each W fragment feeds 2 wmmas (8 FLOP/byte from cache).
// 4 waves/block cover 128 columns.  grid = (M/32, N/128), block = 128.
// ---------------------------------------------------------------------------
template <int PAIRS, bool HAS_BIAS, bool HAS_RESID, bool GELU>
__global__ __launch_bounds__(128) void gemm_wmma_kernel(
    const float* __restrict__ A0, const float* __restrict__ W0,
    const float* __restrict__ A1, const float* __restrict__ W1,
    const float* __restrict__ A2, const float* __restrict__ W2,
    const float* __restrict__ bias, const float* __restrict__ resid,
    float* __restrict__ out, int N, int K, int ldw) {
  const int lane = threadIdx.x & 31;
  const int wave = threadIdx.x >> 5;
  const int mt = blockIdx.x * 32;
  const int nt = blockIdx.y * 128 + wave * 32;
  const int lm = lane & 15;
  const int kk = (lane >> 4) << 1;

  v8f acc[2][2] = {};

  const float* As[3] = {A0, A1, A2};
  const float* Ws[3] = {W0, W1, W2};
#pragma unroll
  for (int p = 0; p < PAIRS; ++p) {
    const float* a0 = As[p] + (size_t)(mt + lm) * K + kk;
    const float* a1 = As[p] + (size_t)(mt + 16 + lm) * K + kk;
    const float* w0 = Ws[p] + (size_t)(nt + lm) * ldw + kk;
    const float* w1 = Ws[p] + (size_t)(nt + 16 + lm) * ldw + kk;
#pragma unroll 4
    for (int k = 0; k < K; k += 4) {
      const v2f av0 = *(const v2f*)(a0 + k);
      const v2f av1 = *(const v2f*)(a1 + k);
      const v2f wv0 = *(const v2f*)(w0 + k);
      const v2f wv1 = *(const v2f*)(w1 + k);
      acc[0][0] = __builtin_amdgcn_wmma_f32_16x16x4_f32(false, av0, false, wv0,
                                                        (short)0, acc[0][0],
                                                        false, false);
      acc[0][1] = __builtin_amdgcn_wmma_f32_16x16x4_f32(false, av0, false, wv1,
                                                        (short)0, acc[0][1],
                                                        false, false);
      acc[1][0] = __builtin_amdgcn_wmma_f32_16x16x4_f32(false, av1, false, wv0,
                                                        (short)0, acc[1][0],
                                                        false, false);
      acc[1][1] = __builtin_amdgcn_wmma_f32_16x16x4_f32(false, av1, false, wv1,
                                                        (short)0, acc[1][1],
                                                        false, false);
    }
  }

#pragma unroll
  for (int ri = 0; ri < 2; ++ri) {
#pragma unroll
    for (int ci = 0; ci < 2; ++ci) {
      const int col = nt + 16 * ci + lm;
      const float bv = HAS_BIAS ? bias[col] : 0.f;
      const int rbase = mt + 16 * ri + ((lane >> 4) << 3);
#pragma unroll
      for (int i = 0; i < 8; ++i) {
        float v = acc[ri][ci][i] + bv;
        if constexpr (GELU) v = 0.5f * v * (1.f + erff(v * 0.70710678118654752f));
        const size_t idx = (size_t)(rbase + i) * N + col;
        if constexpr (HAS_RESID) v += resid[idx];
        out[idx] = v;
      }
    }
  }
}

// ---------------------------------------------------------------------------
// GNN aggregation exploiting the fixed fan-in structure:
//   u[t] = dinv*sum_j w_j x[src_j] + dinv^2 * x[t],  m[t] = mean_j x[src_j]
// wave per target; lane handles 4 of 128 channels.  grid = T/8, block = 256.
// ---------------------------------------------------------------------------
__global__ __launch_bounds__(256) void gnn_aggregate_kernel(
    const int* __restrict__ n_id, const int* __restrict__ e_id,
    const float* __restrict__ edge_attr, const float* __restrict__ item_emb,
    float* __restrict__ u_out, float* __restrict__ m_out) {
  const int lane = threadIdx.x & 31;
  const int wave = threadIdx.x >> 5;
  const int t = blockIdx.x * 8 + wave;
  const int h0 = lane * 4;
  float4 s0 = {0, 0, 0, 0}, s1 = {0, 0, 0, 0};
  float wsum = 0.f;
  const int ebase = t * FANz;
  for (int j = 0; j < FANz; ++j) {
    const int e = ebase + j;
    const float w = edge_attr[e_id[e]];
    const int src = n_id[Tz + e];
    const float4 v = *(const float4*)(item_emb + (size_t)src * Hz + h0);
    s0.x += v.x; s0.y += v.y; s0.z += v.z; s0.w += v.w;
    s1.x += w * v.x; s1.y += w * v.y; s1.z += w * v.z; s1.w += w * v.w;
    wsum += w;
  }
  const float dinv = 1.f / sqrtf(wsum + 1.f);
  const float d2 = dinv * dinv;
  const int self = n_id[t];
  const float4 xv = *(const float4*)(item_emb + (size_t)self * Hz + h0);
  float4 u, m;
  u.x = dinv * s1.x + d2 * xv.x; u.y = dinv * s1.y + d2 * xv.y;
  u.z = dinv * s1.z + d2 * xv.z; u.w = dinv * s1.w + d2 * xv.w;
  const float inv_fan = 1.f / (float)FANz;
  m.x = s0.x * inv_fan; m.y = s0.y * inv_fan;
  m.z = s0.z * inv_fan; m.w = s0.w * inv_fan;
  *(float4*)(u_out + (size_t)t * Hz + h0) = u;
  *(float4*)(m_out + (size_t)t * Hz + h0) = m;
}

// ---------------------------------------------------------------------------
// Row L2 normalize in place (F.normalize).  wave per row.
// ---------------------------------------------------------------------------
__global__ __launch_bounds__(256) void l2norm_kernel(float* __restrict__ data) {
  const int lane = threadIdx.x & 31;
  const int row = blockIdx.x * 8 + (threadIdx.x >> 5);
  const int h0 = lane * 4;
  float4 v = *(const float4*)(data + (size_t)row * Hz + h0);
  const float ss = wave_sum(v.x * v.x + v.y * v.y + v.z * v.z + v.w * v.w);
  const float s = 1.f / fmaxf(sqrtf(ss), 1e-12f);
  v.x *= s; v.y *= s; v.z *= s; v.w *= s;
  *(float4*)(data + (size_t)row * Hz + h0) = v;
}

// ---------------------------------------------------------------------------
// Row LayerNorm; optional gather index (for item_emb[inputs]).  wave per row.
// ---------------------------------------------------------------------------
__global__ __launch_bounds__(256) void row_ln_kernel(
    const float* __restrict__ base, const int* __restrict__ gidx,
    const float* __restrict__ g, const float* __restrict__ b,
    float* __restrict__ out, float eps) {
  const int lane = threadIdx.x & 31;
  const int row = blockIdx.x * 8 + (threadIdx.x >> 5);
  const float* src =
      gidx ? base + (size_t)gidx[row] * Hz : base + (size_t)row * Hz;
  const int h0 = lane * 4;
  const float4 v = *(const float4*)(src + h0);
  const float mean = wave_sum(v.x + v.y + v.z + v.w) * (1.f / Hz);
  const float dx = v.x - mean, dy = v.y - mean, dz = v.z - mean, dw = v.w - mean;
  const float var = wave_sum(dx * dx + dy * dy + dz * dz + dw * dw) * (1.f / Hz);
  const float inv = 1.f / sqrtf(var + eps);
  float4 o;
  o.x = dx * inv * g[h0 + 0] + b[h0 + 0];
  o.y = dy * inv * g[h0 + 1] + b[h0 + 1];
  o.z = dz * inv * g[h0 + 2] + b[h0 + 2];
  o.w = dw * inv * g[h0 + 3] + b[h0 + 3];
  *(float4*)(out + (size_t)row * Hz + h0) = o;
}

// ---------------------------------------------------------------------------
// Causal MHA for one (batch, head).  L=50, hd=32 — tiny, VALU is fine.
// grid = (B, NH), block = 64.
// ---------------------------------------------------------------------------
__global__ __launch_bounds__(64) void attention_kernel(
    const float* __restrict__ qkv, float* __restrict__ o) {
  const int b = blockIdx.x, h = blockIdx.y;
  __shared__ float qs[Lz * HDz], ks[Lz * HDz], vs[Lz * HDz];
  for (int idx = threadIdx.x; idx < Lz * HDz; idx += blockDim.x) {
    const int l = idx / HDz, d = idx % HDz;
    const float* rowp = qkv + (size_t)(b * Lz + l) * (3 * Hz);
    qs[idx] = rowp[h * HDz + d];
    ks[idx] = rowp[Hz + h * HDz + d];
    vs[idx] = rowp[2 * Hz + h * HDz + d];
  }
  __syncthreads();
  const int l = threadIdx.x;
  if (l < Lz) {
    float sc[Lz];
    float mx = -3.0e38f;
    for (int j = 0; j <= l; ++j) {
      float s = 0.f;
      for (int d = 0; d < HDz; ++d) s += qs[l * HDz + d] * ks[j * HDz + d];
      s *= 0.17677669529663687f;  // 1/sqrt(32)
      sc[j] = s;
      mx = fmaxf(mx, s);
    }
    float den = 0.f;
    for (int j = 0; j <= l; ++j) {
      sc[j] = expf(sc[j] - mx);
      den += sc[j];
    }
    const float inv = 1.f / den;
    for (int d = 0; d < HDz; ++d) {
      float acc = 0.f;
      for (int j = 0; j <= l; ++j) acc += sc[j] * vs[j * HDz + d];
      o[(size_t)(b * Lz + l) * Hz + h * HDz + d] = acc * inv;
    }
  }
}

// ---------------------------------------------------------------------------
// Gate: g[row] *= sigmoid(g[row].gate_item + user_emb[uid[b]].gate_user[:,l])
// wave per (b,l) row, in place.
// ---------------------------------------------------------------------------
__global__ __launch_bounds__(256) void gate_kernel(
    float* __restrict__ g, const float* __restrict__ gate_item,
    const float* __restrict__ gate_user, const float* __restrict__ user_emb,
    const int* __restrict__ user_ids) {
  const int lane = threadIdx.x & 31;
  const int row = blockIdx.x * 8 + (threadIdx.x >> 5);
  const int bb = row / Lz, l = row % Lz;
  const float* ue = user_emb + (size_t)user_ids[bb] * Hz;
  const int h0 = lane * 4;
  float4 gv = *(const float4*)(g + (size_t)row * Hz + h0);
  float acc = 0.f;
#pragma unroll
  for (int j = 0; j < 4; ++j) {
    const int hh = h0 + j;
    const float ge = (j == 0 ? gv.x : j == 1 ? gv.y : j == 2 ? gv.z : gv.w);
    acc += ge * gate_item[hh] + ue[hh] * gate_user[(size_t)hh * Lz + l];
  }
  acc = wave_sum(acc);
  const float gs = 1.f / (1.f + expf(-acc));
  gv.x *= gs; gv.y *= gs; gv.z *= gs; gv.w *= gs;
  *(float4*)(g + (size_t)row * Hz + h0) = gv;
}

// ---------------------------------------------------------------------------
extern "C" void kernel_launch(void* const* d_in, const int* in_sizes, int n_in,
                              void* d_out, int out_size, void* d_ws,
                              size_t ws_size, hipStream_t stream) {
  (void)in_sizes; (void)n_in; (void)out_size; (void)ws_size;
  const int*   user_ids  = (const int*)d_in[0];
  const int*   inputs    = (const int*)d_in[1];
  const int*   n_id_a    = (const int*)d_in[2];
  const int*   e_id_a    = (const int*)d_in[4];
  const int*   n_id_b    = (const int*)d_in[5];
  const int*   e_id_b    = (const int*)d_in[7];
  const float* edge_attr = (const float*)d_in[8];
  const float* item_emb  = (const float*)d_in[9];
  const float* user_emb  = (const float*)d_in[10];
  const float* ln_g      = (const float*)d_in[11];
  const float* ln_b      = (const float*)d_in[12];
  const float* gcn_w     = (const float*)d_in[13];
  const float* gcn_b     = (const float*)d_in[14];
  const float* sage_wl   = (const float*)d_in[15];
  const float* sage_bl   = (const float*)d_in[16];
  const float* sage_wr   = (const float*)d_in[17];
  const float* gate_item = (const float*)d_in[18];
  const float* gate_user = (const float*)d_in[19];
  const float* wt        = (const float*)d_in[20];
  const float* qkv_w     = (const float*)d_in[21];
  const float* qkv_b     = (const float*)d_in[22];
  const float* out_w     = (const float*)d_in[23];
  const float* out_b     = (const float*)d_in[24];
  const float* ln1_g     = (const float*)d_in[25];
  const float* ln1_b     = (const float*)d_in[26];
  const float* ff_w1     = (const float*)d_in[27];
  const float* ff_b1     = (const float*)d_in[28];
  const float* ff_w2     = (const float*)d_in[29];
  const float* ff_b2     = (const float*)d_in[30];
  const float* ln2_g     = (const float*)d_in[31];
  const float* ln2_b     = (const float*)d_in[32];

  float* ws = (float*)d_ws;
  const size_t TH = (size_t)Tz * Hz;
  float* f_agg = ws;                 // 2*TH  (u rows, then m rows)
  float* f_h   = f_agg + 2 * TH;     // 2*TH  (hu rows, then hm rows)
  float* f_ga  = f_h + 2 * TH;       // TH
  float* f_gb  = f_ga + TH;          // TH
  float* f_x   = f_gb + TH;          // TH    encoder state
  float* f_y   = f_x + TH;           // TH    pre-LN temp
  float* f_qkv = f_y + TH;           // 3*TH
  float* f_o   = f_qkv + 3 * TH;     // TH
  float* f_f1  = f_agg;              // reuse 4*TH after GNN phase

  const dim3 blk256(256);
  const dim3 gblk(128);
  const float* NF = nullptr;

  // ---- GNN on both graphs --------------------------------------------------
  for (int gph = 0; gph < 2; ++gph) {
    const int* n_id = gph ? n_id_b : n_id_a;
    const int* e_id = gph ? e_id_b : e_id_a;
    float* gout = gph ? f_gb : f_ga;
    gnn_aggregate_kernel<<<Tz / 8, blk256, 0, stream>>>(
        n_id, e_id, edge_attr, item_emb, f_agg, f_agg + TH);
    // [u;m] @ gcn_w^T + gcn_b  (M = 2T)
    gemm_wmma_kernel<1, true, false, false>
        <<<dim3(2 * Tz / 32, 1), gblk, 0, stream>>>(
            f_agg, gcn_w, NF, NF, NF, NF, gcn_b, NF, f_h, Hz, Hz, Hz);
    // hm @ sage_wl^T + sage_bl + hu @ sage_wr^T
    gemm_wmma_kernel<2, true, false, false>
        <<<dim3(Tz / 32, 1), gblk, 0, stream>>>(
            f_h + TH, sage_wl, f_h, sage_wr, NF, NF, sage_bl, NF, gout, Hz, Hz,
            Hz);
    l2norm_kernel<<<Tz / 8, blk256, 0, stream>>>(gout);
  }

  // ---- Embedding LayerNorm -------------------------------------------------
  row_ln_kernel<<<Tz / 8, blk256, 0, stream>>>(item_emb, inputs, ln_g, ln_b,
                                               f_x, 1e-12f);

  // ---- Transformer encoder (post-LN, causal) -------------------------------
  for (int i = 0; i < NLz; ++i) {
    gemm_wmma_kernel<1, true, false, false>
        <<<dim3(Tz / 32, 3), gblk, 0, stream>>>(
            f_x, qkv_w + (size_t)i * 3 * Hz * Hz, NF, NF, NF, NF,
            qkv_b + i * 3 * Hz, NF, f_qkv, 3 * Hz, Hz, Hz);
    attention_kernel<<<dim3(Bz, NHz), dim3(64), 0, stream>>>(f_qkv, f_o);
    gemm_wmma_kernel<1, true, true, false>
        <<<dim3(Tz / 32, 1), gblk, 0, stream>>>(
            f_o, out_w + (size_t)i * Hz * Hz, NF, NF, NF, NF, out_b + i * Hz,
            f_x, f_y, Hz, Hz, Hz);
    row_ln_kernel<<<Tz / 8, blk256, 0, stream>>>(f_y, nullptr, ln1_g + i * Hz,
                                                 ln1_b + i * Hz, f_x, 1e-5f);
    gemm_wmma_kernel<1, true, false, true>
        <<<dim3(Tz / 32, 4), gblk, 0, stream>>>(
            f_x, ff_w1 + (size_t)i * 4 * Hz * Hz, NF, NF, NF, NF,
            ff_b1 + i * 4 * Hz, NF, f_f1, 4 * Hz, Hz, Hz);
    gemm_wmma_kernel<1, true, true, false>
        <<<dim3(Tz / 32, 1), gblk, 0, stream>>>(
            f_f1, ff_w2 + (size_t)i * Hz * 4 * Hz, NF, NF, NF, NF,
            ff_b2 + i * Hz, f_x, f_y, Hz, 4 * Hz, 4 * Hz);
    row_ln_kernel<<<Tz / 8, blk256, 0, stream>>>(f_y, nullptr, ln2_g + i * Hz,
                                                 ln2_b + i * Hz, f_x, 1e-5f);
  }

  // ---- Gating (in place on ga/gb) ------------------------------------------
  gate_kernel<<<Tz / 8, blk256, 0, stream>>>(f_ga, gate_item, gate_user,
                                             user_emb, user_ids);
  gate_kernel<<<Tz / 8, blk256, 0, stream>>>(f_gb, gate_item, gate_user,
                                             user_emb, user_ids);

  // ---- Final: concat([seq, usa, usb]) @ wt^T  (three K=128 slices of wt) ----
  gemm_wmma_kernel<3, false, false, false>
      <<<dim3(Tz / 32, 1), gblk, 0, stream>>>(
          f_x, wt, f_ga, wt + Hz, f_gb, wt + 2 * Hz, NF, NF, (float*)d_out, Hz,
          Hz, 3 * Hz);
}